// GResBlock_71305047048353
// MI455X (gfx1250) — compile-verified
//
#include <hip/hip_runtime.h>

#define D_DIM 96
#define NN 50000
#define NE 800000
#define PSTR 112   // K-pair row stride in v2f units: 2*112 = 224 dwords == 32 (mod 64)
                   // -> the two half-wave K-pair rows hit disjoint 32-bank windows

typedef __attribute__((ext_vector_type(2))) float v2f;
typedef __attribute__((ext_vector_type(8))) float v8f;

// ---------------------------------------------------------------------------
// Dual GEMM: support = act(X) @ W ; acc = act(X) @ Wl + bias
// One wave -> one 16-row tile, all 96 output columns, fp32 WMMA 16x16x4.
// Weights staged in LDS in K-pair-major layout so every B-fragment is a
// single ds_load_b64 into an even-aligned VGPR pair (no register shuffles).
// ---------------------------------------------------------------------------
template <bool RELU_IN>
__global__ __launch_bounds__(256) void gemm_dual_kernel(
    const float* __restrict__ X, const float* __restrict__ W,
    const float* __restrict__ Wl, const float* __restrict__ bias,
    float* __restrict__ support, float* __restrict__ acc) {
  __shared__ v2f sW[48 * PSTR];    // sW[kpair*PSTR + n] = {W[2kp][n], W[2kp+1][n]}
  __shared__ v2f sWl[48 * PSTR];
  {
    float* fW = reinterpret_cast<float*>(sW);
    float* fWl = reinterpret_cast<float*>(sWl);
    for (int i = threadIdx.x; i < D_DIM * D_DIM; i += 256) {
      const int k = i / D_DIM, n = i - k * D_DIM;
      const int slot = ((k >> 1) * PSTR + n) * 2 + (k & 1);
      fW[slot] = W[i];
      fWl[slot] = Wl[i];
    }
  }
  __syncthreads();

  const int wave = threadIdx.x >> 5;
  const int lane = threadIdx.x & 31;
  const int rowTile = blockIdx.x * 8 + wave;       // 16 rows per wave
  if (rowTile >= NN / 16) return;                  // wave-uniform: EXEC stays all-1s

  const int mRow = rowTile * 16 + (lane & 15);     // A: both lane halves carry M=0..15
  const int kpHalf = lane >> 4;                    // half-wave K-pair select (0 or 1)
  const int nLane = lane & 15;

  v8f accS[6] = {};
  v8f accL[6] = {};

  const float* xrow = X + (size_t)mRow * D_DIM;

#pragma unroll 4
  for (int ks = 0; ks < 24; ++ks) {                // K = 96 = 24 * 4
    const int kp = ks * 2 + kpHalf;                // this lane's K-pair index
    v2f a = *reinterpret_cast<const v2f*>(xrow + kp * 2);  // K = 2kp, 2kp+1
    if (RELU_IN) {
      a.x = fmaxf(a.x, 0.0f);
      a.y = fmaxf(a.y, 0.0f);
    }
    const v2f* rowS = sW + kp * PSTR;
    const v2f* rowL = sWl + kp * PSTR;
#pragma unroll
    for (int j = 0; j < 6; ++j) {                  // 6 N-tiles of 16 columns
      const v2f bS = rowS[j * 16 + nLane];         // single ds_load_b64
      const v2f bL = rowL[j * 16 + nLane];
      accS[j] = __builtin_amdgcn_wmma_f32_16x16x4_f32(false, a, false, bS,
                                                      (short)0, accS[j], false, false);
      accL[j] = __builtin_amdgcn_wmma_f32_16x16x4_f32(false, a, false, bL,
                                                      (short)0, accL[j], false, false);
    }
  }

  // C/D layout: VGPR v -> row rowBase+v; lanes 0-15 rows 0-7, lanes 16-31 rows 8-15
  const int rowBase = rowTile * 16 + ((lane >> 4) << 3);
#pragma unroll
  for (int j = 0; j < 6; ++j) {
    const int n = j * 16 + nLane;
    const float bv = bias[n];
#pragma unroll
    for (int v = 0; v < 8; ++v) {
      const size_t idx = (size_t)(rowBase + v) * D_DIM + n;
      support[idx] = accS[j][v];
      acc[idx] = accL[j][v] + bv;                  // fold bias into loop-term
    }
  }
}

// ---------------------------------------------------------------------------
// Edge scatter: acc[dst] += support[src] * w   (24 float4 lanes per edge)
// support table (19.2 MB) is L2-resident; atomics are hardware f32 adds.
// ---------------------------------------------------------------------------
__global__ __launch_bounds__(256) void scatter_kernel(
    const float* __restrict__ support, const float* __restrict__ ew,
    const int* __restrict__ esrc, const int* __restrict__ edst,
    float* __restrict__ acc) {
  const long long tid = (long long)blockIdx.x * 256 + threadIdx.x;
  const int e = (int)(tid / 24);
  if (e >= NE) return;
  const int c = (int)(tid - (long long)e * 24) * 4;
  const int s = esrc[e];
  const int d = edst[e];
  const float w = ew[e];
  const float4 v = *reinterpret_cast<const float4*>(support + (size_t)s * D_DIM + c);
  float* ap = acc + (size_t)d * D_DIM + c;
  unsafeAtomicAdd(ap + 0, v.x * w);   // global_atomic_add_f32, no return
  unsafeAtomicAdd(ap + 1, v.y * w);
  unsafeAtomicAdd(ap + 2, v.z * w);
  unsafeAtomicAdd(ap + 3, v.w * w);
}

// ---------------------------------------------------------------------------
// Final fuse: out = (x + relu(out)) * 0.5
// ---------------------------------------------------------------------------
__global__ __launch_bounds__(256) void final_kernel(const float* __restrict__ x,
                                                    float* __restrict__ out) {
  const int i = blockIdx.x * 256 + threadIdx.x;
  if (i >= NN * D_DIM / 4) return;
  const float4 xv = reinterpret_cast<const float4*>(x)[i];
  float4 h = reinterpret_cast<float4*>(out)[i];
  h.x = (xv.x + fmaxf(h.x, 0.0f)) * 0.5f;
  h.y = (xv.y + fmaxf(h.y, 0.0f)) * 0.5f;
  h.z = (xv.z + fmaxf(h.z, 0.0f)) * 0.5f;
  h.w = (xv.w + fmaxf(h.w, 0.0f)) * 0.5f;
  reinterpret_cast<float4*>(out)[i] = h;
}

extern "C" void kernel_launch(void* const* d_in, const int* in_sizes, int n_in,
                              void* d_out, int out_size, void* d_ws, size_t ws_size,
                              hipStream_t stream) {
  const float* x  = (const float*)d_in[0];
  const float* W1 = (const float*)d_in[1];
  const float* W1l = (const float*)d_in[2];
  const float* b1 = (const float*)d_in[3];
  const float* W2 = (const float*)d_in[4];
  const float* W2l = (const float*)d_in[5];
  const float* b2 = (const float*)d_in[6];
  const float* ew = (const float*)d_in[7];
  const int* esrc = (const int*)d_in[8];
  const int* edst = (const int*)d_in[9];
  float* out = (float*)d_out;

  float* ws0 = (float*)d_ws;                  // support table  [N, D]
  float* ws1 = ws0 + (size_t)NN * D_DIM;      // pre-relu layer-1 accumulator [N, D]

  const dim3 blk(256);
  const int gemmGrid = (NN / 16 + 7) / 8;                        // 391 blocks (8 waves ea.)
  const long long sThreads = (long long)NE * 24;
  const int sGrid = (int)((sThreads + 255) / 256);               // 75000 blocks
  const int fGrid = (NN * D_DIM / 4 + 255) / 256;

  // Layer 1: support1 = x@W1 ; ws1 = x@W1_loop + b1 ; ws1 += scatter(support1)
  gemm_dual_kernel<false><<<gemmGrid, blk, 0, stream>>>(x, W1, W1l, b1, ws0, ws1);
  scatter_kernel<<<sGrid, blk, 0, stream>>>(ws0, ew, esrc, edst, ws1);

  // Layer 2: A = relu(ws1) fused into GEMM load; accumulate loop-term into d_out
  gemm_dual_kernel<true><<<gemmGrid, blk, 0, stream>>>(ws1, W2, W2l, b2, ws0, out);
  scatter_kernel<<<sGrid, blk, 0, stream>>>(ws0, ew, esrc, edst, out);

  // out = (x + relu(out)) * 0.5
  final_kernel<<<fGrid, blk, 0, stream>>>(x, out);
}